// DenSparseMatrix_80049600463346
// MI455X (gfx1250) — compile-verified
//
#include <hip/hip_runtime.h>

// Problem dims (fixed by the reference).
#define I_DIM 65536
#define O_DIM 65536
#define M_DIM 32
#define B_DIM 32
#define ITERS 4   // rows of `i` per wave in the main kernel

typedef __attribute__((ext_vector_type(2))) float v2f;  // A/B for wmma f32 16x16x4 (2 VGPRs)
typedef __attribute__((ext_vector_type(8))) float v8f;  // C/D (8 VGPRs)

// ---------------------------------------------------------------------------
// Kernel 1: transpose x (B=32, I) -> xT (I, 32) so the main kernel's per-row
// loads of x[:, i] are a single coalesced 128B read.
// ---------------------------------------------------------------------------
__global__ void xpose_x_kernel(const float* __restrict__ x, float* __restrict__ xT) {
  __shared__ float tile[32][33];
  const int i0 = blockIdx.x * 32;
  const int tx = threadIdx.x, ty = threadIdx.y;
#pragma unroll
  for (int b = ty; b < 32; b += 8)
    tile[b][tx] = x[(size_t)b * I_DIM + i0 + tx];           // coalesced along I
  __syncthreads();
#pragma unroll
  for (int r = ty; r < 32; r += 8)
    xT[(size_t)(i0 + r) * 32 + tx] = tile[tx][r];           // coalesced along B
}

// ---------------------------------------------------------------------------
// Kernel 2: zero the (O, 32) f32 accumulator (8 MB, L2-resident).
// ---------------------------------------------------------------------------
__global__ void zero_kernel(float4* __restrict__ p) {
  p[(size_t)blockIdx.x * blockDim.x + threadIdx.x] = make_float4(0.f, 0.f, 0.f, 0.f);
}

// ---------------------------------------------------------------------------
// Kernel 3: main. One wave32 per group of ITERS rows `i`. For each i:
//   load w[i,0..31], map[i,0..31], xT[i,0..31] (all coalesced 128B),
//   compute the 32x32 outer product w[i,:] (x) x[:,i] with four
//   V_WMMA_F32_16X16X4_F32 (exact f32 math: K columns 1..3 are zero),
//   then scatter each row m to outT[map[i,m], :] with wave-wide
//   global_atomic_add_f32 in two contiguous 64B segments.
// outT is (O, 32): the 32 b-values of one output row are contiguous, so the
// random-index scatter is coalesced and stays in the 192MB L2. All offsets
// are unsigned 32-bit so the atomics use the SGPR-base + 32-bit-VGPR-offset
// addressing mode (no per-lane 64-bit address math).
// ---------------------------------------------------------------------------
__global__ void scatter_wmma_kernel(const float* __restrict__ fw,
                                    const int*   __restrict__ mapv,
                                    const float* __restrict__ xT,
                                    float*       __restrict__ outT) {
  const int lane = threadIdx.x & 31;
  const int wave = (int)((blockIdx.x * blockDim.x + threadIdx.x) >> 5);
  const bool lo  = lane < 16;
  const unsigned lane15 = (unsigned)(lane & 15);
  const v8f cz = {};

#pragma unroll
  for (int k = 0; k < ITERS; ++k) {
    const size_t i   = (size_t)wave * ITERS + k;
    const size_t off = i * 32 + lane;

    const float wv = fw[off];      // w[i, m=lane]
    const float xv = xT[off];      // x[b=lane, i]
    const int   mp = mapv[off];    // map[i, m=lane]

    // Broadcast the upper-half values down to lanes 0..15 for the second tiles.
    const float wh = __shfl(wv, (lane & 15) + 16, 32);
    const float xh = __shfl(xv, (lane & 15) + 16, 32);

    // A (16x4, M in lanes 0..15, K=0 column holds w; K=1..3 zero)
    const v2f a0 = { lo ? wv : 0.0f, 0.0f };   // m = 0..15
    const v2f a1 = { lo ? wh : 0.0f, 0.0f };   // m = 16..31
    // B (4x16, N in lanes 0..15, K=0 row holds x; K=1..3 zero)
    const v2f b0 = { lo ? xv : 0.0f, 0.0f };   // b = 0..15
    const v2f b1 = { lo ? xh : 0.0f, 0.0f };   // b = 16..31

    // D = A x B + 0 : four 16x16 tiles of the 32x32 outer product.
    const v8f c00 = __builtin_amdgcn_wmma_f32_16x16x4_f32(false, a0, false, b0, (short)0, cz, false, false);
    const v8f c01 = __builtin_amdgcn_wmma_f32_16x16x4_f32(false, a0, false, b1, (short)0, cz, false, false);
    const v8f c10 = __builtin_amdgcn_wmma_f32_16x16x4_f32(false, a1, false, b0, (short)0, cz, false, false);
    const v8f c11 = __builtin_amdgcn_wmma_f32_16x16x4_f32(false, a1, false, b1, (short)0, cz, false, false);

    // C/D layout: VGPR j holds row M=j (lanes 0..15) / M=j+8 (lanes 16..31),
    // column N = lane&15. Scatter row M to outT[map[i,M], :].
#pragma unroll
    for (int j = 0; j < 8; ++j) {
      const int srcA = j + (lo ? 0 : 8);              // m for tile rows 0..15
      // 32-bit element offsets into the 2M-element outT (fits in u32 easily).
      const unsigned e0 = (unsigned)__shfl(mp, srcA, 32)      * 32u + lane15;
      const unsigned e1 = (unsigned)__shfl(mp, srcA + 16, 32) * 32u + lane15;
      atomicAdd(outT + e0,      c00[j]);   // b = 0..15
      atomicAdd(outT + e0 + 16, c01[j]);   // b = 16..31
      atomicAdd(outT + e1,      c10[j]);
      atomicAdd(outT + e1 + 16, c11[j]);
    }
  }
}

// ---------------------------------------------------------------------------
// Kernel 4: transpose outT (O, 32) -> out (B=32, O).
// ---------------------------------------------------------------------------
__global__ void xpose_out_kernel(const float* __restrict__ outT, float* __restrict__ out) {
  __shared__ float tile[32][33];
  const int o0 = blockIdx.x * 32;
  const int tx = threadIdx.x, ty = threadIdx.y;
#pragma unroll
  for (int r = ty; r < 32; r += 8)
    tile[r][tx] = outT[(size_t)(o0 + r) * 32 + tx];         // coalesced rows
  __syncthreads();
#pragma unroll
  for (int b = ty; b < 32; b += 8)
    out[(size_t)b * O_DIM + o0 + tx] = tile[tx][b];         // coalesced along O
}

// ---------------------------------------------------------------------------
extern "C" void kernel_launch(void* const* d_in, const int* in_sizes, int n_in,
                              void* d_out, int out_size, void* d_ws, size_t ws_size,
                              hipStream_t stream) {
  const float* x    = (const float*)d_in[0];   // (B, I) f32
  const float* fw   = (const float*)d_in[1];   // (I, M) f32
  const int*   mapv = (const int*)  d_in[2];   // (I, M) int
  float* out = (float*)d_out;                  // (B, O) f32

  float* xT   = (float*)d_ws;                                   // (I, 32) : 8 MB
  float* outT = (float*)((char*)d_ws + (size_t)I_DIM * 32 * 4); // (O, 32) : 8 MB

  dim3 tb(32, 8);
  xpose_x_kernel<<<I_DIM / 32, tb, 0, stream>>>(x, xT);

  zero_kernel<<<(O_DIM * B_DIM / 4) / 256, 256, 0, stream>>>((float4*)outT);

  // 65536 rows / ITERS per wave / 8 waves per block
  const int blocks = I_DIM / (ITERS * 8);
  scatter_wmma_kernel<<<blocks, 256, 0, stream>>>(fw, mapv, xT, outT);

  xpose_out_kernel<<<O_DIM / 32, tb, 0, stream>>>(outT, out);
}